// MessageOperatorInput_77833397338252
// MI455X (gfx1250) — compile-verified
//
#include <hip/hip_runtime.h>
#include <hip/hip_bf16.h>

typedef float v2f __attribute__((ext_vector_type(2)));
typedef float v4f __attribute__((ext_vector_type(4)));
typedef float v8f __attribute__((ext_vector_type(8)));

#define OUT_DIM 64
#define IN_DIM 128

// ---------------------------------------------------------------- utilities
__global__ void zero_i32_kernel(int* __restrict__ p, int n) {
    int i = blockIdx.x * 256 + threadIdx.x;
    if (i < n) p[i] = 0;
}

__global__ void hist_kernel(const int* __restrict__ eidx, int* __restrict__ counts, int E) {
    int i = blockIdx.x * 256 + threadIdx.x;
    if (i < E) atomicAdd(&counts[eidx[i]], 1);
}

// Single-block exclusive scan over `n` (<= a few 10^4) counts -> offsets & cursors.
__global__ void scan_kernel(const int* __restrict__ counts,
                            int* __restrict__ offsets,
                            int* __restrict__ cursors, int n) {
    __shared__ int tmp[1024];
    int tid = threadIdx.x;
    int carry = 0;
    for (int base = 0; base < n; base += 1024) {
        int i = base + tid;
        int v = (i < n) ? counts[i] : 0;
        tmp[tid] = v;
        __syncthreads();
        // Hillis-Steele inclusive scan
        for (int off = 1; off < 1024; off <<= 1) {
            int t = (tid >= off) ? tmp[tid - off] : 0;
            __syncthreads();
            tmp[tid] += t;
            __syncthreads();
        }
        int incl  = tmp[tid];
        int total = tmp[1023];
        if (i < n) {
            int excl = carry + incl - v;
            offsets[i] = excl;
            cursors[i] = excl;
        }
        carry += total;
        __syncthreads(); // protect tmp before next chunk overwrites it
    }
}

__global__ void scatter_kernel(const int* __restrict__ eidx,
                               int* __restrict__ cursors,
                               int* __restrict__ bins, int E) {
    int i = blockIdx.x * 256 + threadIdx.x;
    if (i < E) {
        int p = eidx[i];
        int pos = atomicAdd(&cursors[p], 1);
        bins[pos] = i;
    }
}

// ------------------------------------------------- grouped WMMA GEMM kernel
// One workgroup (4 waves, 128 threads) per pool index p.
// Wave w owns rows [16w, 16w+16) of W_p; loops over groups of 16 edges that
// map to p, computing D(16x16) += A(16x4, W rows) x B(4x16, gathered x cols)
// via V_WMMA_F32_16X16X4_F32 accumulated across 32 k-chunks.
__global__ void edge_gemm_grouped(const float* __restrict__ nodes,
                                  const float* __restrict__ core,
                                  const float* __restrict__ bias,
                                  const int* __restrict__ offsets,
                                  const int* __restrict__ counts,
                                  const int* __restrict__ bins,
                                  float* __restrict__ out) {
    int p = blockIdx.x;
    int start = offsets[p];
    int cnt = counts[p];
    if (cnt == 0) return;               // uniform per block
    int end = start + cnt;

    int tid  = threadIdx.x;
    int lane = tid & 31;
    int wave = tid >> 5;                // 0..3 : 16-row chunk of W
    int n    = lane & 15;               // A: row-in-chunk, B/D: column
    int half = lane >> 4;               // K sub-offset (A/B), row half (D)

    // A fragment source: W_p row (16*wave + n), elements 4c + 2*half + {0,1}
    const float* Wrow = core + (size_t)p * (OUT_DIM * IN_DIM)
                             + (size_t)(wave * 16 + n) * IN_DIM + 2 * half;
    // Bias: lane covers rows 16*wave + 8*half + [0,8)
    const float* Brow = bias + (size_t)p * OUT_DIM + wave * 16 + 8 * half;

    for (int g = 0; start + g * 16 < end; ++g) {
        int ci = start + g * 16 + n;
        bool valid = ci < end;
        int eid = bins[valid ? ci : start];      // dup col 0 for tail lanes
        const float* Xp = nodes + (size_t)eid * IN_DIM + 2 * half;

        v8f acc = {};
        #pragma unroll
        for (int c = 0; c < 32; ++c) {
            v2f a = *(const v2f*)(Wrow + 4 * c);
            v2f b = *(const v2f*)(Xp + 4 * c);
            acc = __builtin_amdgcn_wmma_f32_16x16x4_f32(
                false, a, false, b, (short)0, acc, false, false);
        }

        if (valid) {
            float* Op = out + (size_t)eid * OUT_DIM + wave * 16 + 8 * half;
            const float* ac = (const float*)&acc;
            v4f o0, o1;
            #pragma unroll
            for (int r = 0; r < 4; ++r) {
                o0[r] = fmaxf(ac[r]     + Brow[r],     0.0f);
                o1[r] = fmaxf(ac[r + 4] + Brow[r + 4], 0.0f);
            }
            *(v4f*)(Op)     = o0;      // 32B-aligned: eid*64 + 16w + 8*half
            *(v4f*)(Op + 4) = o1;
        }
    }
}

// ------------------------------------- fallback: wave-per-edge WMMA matvec
// Used only if d_ws is too small for binning scratch. B columns all replicate
// the edge's x vector; only lanes with n==0 store (columns identical).
__global__ void edge_gemv_naive(const float* __restrict__ nodes,
                                const int* __restrict__ eidx,
                                const float* __restrict__ core,
                                const float* __restrict__ bias,
                                float* __restrict__ out, int E) {
    int wave = threadIdx.x >> 5;
    int e = blockIdx.x * 8 + wave;
    if (e >= E) return;                 // uniform per wave
    int lane = threadIdx.x & 31;
    int n = lane & 15, half = lane >> 4;
    int p = eidx[e];

    const float* W0 = core + (size_t)p * (OUT_DIM * IN_DIM)
                           + (size_t)n * IN_DIM + 2 * half;
    const float* Xp = nodes + (size_t)e * IN_DIM + 2 * half;

    v8f a0 = {}, a1 = {}, a2 = {}, a3 = {};
    #pragma unroll
    for (int c = 0; c < 32; ++c) {
        v2f b = *(const v2f*)(Xp + 4 * c);
        v2f w;
        w = *(const v2f*)(W0 + 4 * c);
        a0 = __builtin_amdgcn_wmma_f32_16x16x4_f32(false, w, false, b, (short)0, a0, false, false);
        w = *(const v2f*)(W0 + 16 * IN_DIM + 4 * c);
        a1 = __builtin_amdgcn_wmma_f32_16x16x4_f32(false, w, false, b, (short)0, a1, false, false);
        w = *(const v2f*)(W0 + 32 * IN_DIM + 4 * c);
        a2 = __builtin_amdgcn_wmma_f32_16x16x4_f32(false, w, false, b, (short)0, a2, false, false);
        w = *(const v2f*)(W0 + 48 * IN_DIM + 4 * c);
        a3 = __builtin_amdgcn_wmma_f32_16x16x4_f32(false, w, false, b, (short)0, a3, false, false);
    }

    if (n == 0) {
        const float* Bp = bias + (size_t)p * OUT_DIM;
        float* Op = out + (size_t)e * OUT_DIM;
        v8f accs[4] = {a0, a1, a2, a3};
        #pragma unroll
        for (int q = 0; q < 4; ++q) {
            const float* ac = (const float*)&accs[q];
            int m0 = q * 16 + 8 * half;
            #pragma unroll
            for (int r = 0; r < 8; ++r)
                Op[m0 + r] = fmaxf(ac[r] + Bp[m0 + r], 0.0f);
        }
    }
}

// --------------------------------------------------------------- dispatcher
extern "C" void kernel_launch(void* const* d_in, const int* in_sizes, int n_in,
                              void* d_out, int out_size, void* d_ws, size_t ws_size,
                              hipStream_t stream) {
    const float* nodes = (const float*)d_in[0];
    const int*   eidx  = (const int*)d_in[1];
    const float* core  = (const float*)d_in[2];
    const float* bias  = (const float*)d_in[3];
    float* out = (float*)d_out;

    int E    = in_sizes[1];
    int pool = in_sizes[3] / OUT_DIM;

    size_t needed = (size_t)(3 * pool + E) * sizeof(int);
    if (ws_size >= needed && pool > 0) {
        int* counts  = (int*)d_ws;
        int* offsets = counts + pool;
        int* cursors = offsets + pool;
        int* bins    = cursors + pool;

        zero_i32_kernel<<<(pool + 255) / 256, 256, 0, stream>>>(counts, pool);
        hist_kernel<<<(E + 255) / 256, 256, 0, stream>>>(eidx, counts, E);
        scan_kernel<<<1, 1024, 0, stream>>>(counts, offsets, cursors, pool);
        scatter_kernel<<<(E + 255) / 256, 256, 0, stream>>>(eidx, cursors, bins, E);
        edge_gemm_grouped<<<pool, 128, 0, stream>>>(nodes, core, bias,
                                                    offsets, counts, bins, out);
    } else {
        edge_gemv_naive<<<(E + 7) / 8, 256, 0, stream>>>(nodes, eidx, core, bias, out, E);
    }
}